// LinearStateSpaceLayer_43490838839560
// MI455X (gfx1250) — compile-verified
//
#include <hip/hip_runtime.h>

// ============================================================================
// LinearStateSpaceLayer forward for MI455X (gfx1250), chunked-state algorithm.
//
// Roofline: output = 32MB fp32 -> ~1.4us at 23.3 TB/s HBM. Compute after the
// chunked-state reformulation is ~4.3 GFLOP, so the kernel is bandwidth bound
// and we keep everything in fp32 using V_WMMA_F32_16X16X4_F32 (exact-precision
// matrix path; throughput irrelevant at this arithmetic intensity).
// Main kernel stages operands with GLOBAL_LOAD_ASYNC_TO_LDS_B128 (ASYNCcnt),
// overlapping the 80KB tap/C staging with the sequential state recurrence.
// ============================================================================

typedef float v2f __attribute__((ext_vector_type(2)));
typedef float v8f __attribute__((ext_vector_type(8)));

#define STATE 64
#define CH    128
#define SEQ   4096
#define BATCH 16
#define QCH   128          // chunk length
#define NCHK  32           // SEQ / QCH

// workspace offsets (in floats); total ~98.5K floats (~394 KB)
#define OFF_AB 0           // A_bar        64x64
#define OFF_BB 4096        // B_bar        64
#define OFF_VS 4224        // vs[j][n]     128x64   (A_bar^j B_bar, j<128)
#define OFF_KK 12416       // k[s][c]      128x128  (conv kernel taps, s<128)
#define OFF_P  28800       // A_bar^128    64x64
#define OFF_R  32896       // r[(b,m)][n]  512x64   (chunk input injections)
#define OFF_X  65664       // x[m][b][n]   32x16x64 (chunk-boundary states)

static __device__ __forceinline__ v8f wmma_f32_k4(v2f a, v2f b, v8f c) {
  // D(16x16,f32) = A(16x4,f32) * B(4x16,f32) + C
  return __builtin_amdgcn_wmma_f32_16x16x4_f32(false, a, false, b,
                                               (short)0, c, false, false);
}

// async copy of 16 bytes global -> LDS, tracked by ASYNCcnt (per-lane addrs)
static __device__ __forceinline__ void async_ld_b128(unsigned lds_off,
                                                     const void* g) {
  asm volatile("global_load_async_to_lds_b128 %0, %1, off"
               :: "v"(lds_off), "v"((unsigned long long)(__UINTPTR_TYPE__)g)
               : "memory");
}

// ---------------------------------------------------------------------------
// K1: bilinear discretization via Gauss-Jordan on the augmented system
//     (I - 0.5*dt*A) [A_bar | B_bar] = [I + 0.5*dt*A | dt*B]
// ---------------------------------------------------------------------------
__global__ __launch_bounds__(256) void ssm_setup(const float* __restrict__ A,
                                                 const float* __restrict__ Bv,
                                                 const float* __restrict__ dt,
                                                 float* __restrict__ ws) {
  __shared__ float S[64 * 130];     // 64 rows x 129 cols (pad to 130)
  __shared__ float fac[64];
  const int tid = threadIdx.x;
  const float c = 0.5f * dt[0];
  const float dts = dt[0];

  for (int idx = tid; idx < 64 * 129; idx += 256) {
    int r = idx / 129, j = idx % 129;
    float v;
    if (j < 64)        v = (r == j ? 1.0f : 0.0f) - c * A[r * 64 + j];
    else if (j < 128)  v = (r == (j - 64) ? 1.0f : 0.0f) + c * A[r * 64 + (j - 64)];
    else               v = dts * Bv[r];
    S[r * 130 + j] = v;
  }
  __syncthreads();

  for (int p = 0; p < 64; ++p) {
    float inv = 1.0f / S[p * 130 + p];
    __syncthreads();
    for (int j = tid; j < 129; j += 256) S[p * 130 + j] *= inv;
    __syncthreads();
    if (tid < 64) fac[tid] = S[tid * 130 + p];
    __syncthreads();
    for (int idx = tid; idx < 64 * 129; idx += 256) {
      int r = idx / 129, j = idx % 129;
      if (r != p) S[r * 130 + j] -= fac[r] * S[p * 130 + j];
    }
    __syncthreads();
  }

  for (int idx = tid; idx < 4096; idx += 256) {
    int r = idx >> 6, j = idx & 63;
    ws[OFF_AB + idx] = S[r * 130 + 64 + j];
  }
  if (tid < 64) ws[OFF_BB + tid] = S[tid * 130 + 128];
}

// ---------------------------------------------------------------------------
// K2: vs[j] = A_bar^j B_bar for j=0..127 (sequential matvec chain) and
//     P = A_bar^128 via 7 LDS squarings.
// ---------------------------------------------------------------------------
__global__ __launch_bounds__(256) void ssm_vs_pow(float* __restrict__ ws) {
  __shared__ float Pa[64 * 64];
  __shared__ float Pb[64 * 64];
  __shared__ float vcur[64];
  const int tid = threadIdx.x;

  for (int idx = tid; idx < 4096; idx += 256) Pa[idx] = ws[OFF_AB + idx];
  if (tid < 64) vcur[tid] = ws[OFF_BB + tid];
  __syncthreads();
  if (tid < 64) ws[OFF_VS + tid] = vcur[tid];   // vs[0] = B_bar

  for (int j = 1; j < QCH; ++j) {
    float s = 0.0f;
    if (tid < 64) {
      #pragma unroll 8
      for (int k = 0; k < 64; ++k) s += Pa[tid * 64 + k] * vcur[k];
    }
    __syncthreads();
    if (tid < 64) { vcur[tid] = s; ws[OFF_VS + j * 64 + tid] = s; }
    __syncthreads();
  }

  for (int it = 0; it < 7; ++it) {            // A^2,4,8,16,32,64,128
    float* src = (it & 1) ? Pb : Pa;
    float* dst = (it & 1) ? Pa : Pb;
    for (int idx = tid; idx < 4096; idx += 256) {
      int r = idx >> 6, cc = idx & 63;
      float s = 0.0f;
      #pragma unroll 8
      for (int k = 0; k < 64; ++k) s += src[r * 64 + k] * src[k * 64 + cc];
      dst[idx] = s;
    }
    __syncthreads();
  }
  // after 7 squarings the result (A^128) sits in Pb
  for (int idx = tid; idx < 4096; idx += 256) ws[OFF_P + idx] = Pb[idx];
}

// ---------------------------------------------------------------------------
// K3 (WMMA): conv taps k[s][c] = sum_n vs[s][n] * C[c][n]   (128x128x64 GEMM)
// ---------------------------------------------------------------------------
__global__ __launch_bounds__(256) void ssm_kk(const float* __restrict__ C,
                                              float* __restrict__ ws) {
  const int tid  = threadIdx.x;
  const int wave = tid >> 5;
  const int lane = tid & 31;
  const int half = lane >> 4;
  const int lc   = lane & 15;
  const float* vs = ws + OFF_VS;

  for (int tt = wave; tt < 64; tt += 8) {     // 8x8 tiles of 16x16
    const int it = tt >> 3, ct = tt & 7;
    v8f acc = {0.f, 0.f, 0.f, 0.f, 0.f, 0.f, 0.f, 0.f};
    #pragma unroll
    for (int kb = 0; kb < 16; ++kb) {         // K = 64
      const int k0 = kb * 4 + 2 * half;
      const int mr = it * 16 + lc;
      v2f a, b;
      a.x = vs[mr * 64 + k0];
      a.y = vs[mr * 64 + k0 + 1];
      const int nc = ct * 16 + lc;
      b.x = C[nc * 64 + k0];                  // Ct[n][c] = C[c][n]
      b.y = C[nc * 64 + k0 + 1];
      acc = wmma_f32_k4(a, b, acc);
    }
    const int ccol = ct * 16 + lc;
    #pragma unroll
    for (int v = 0; v < 8; ++v) {
      const int srow = it * 16 + half * 8 + v;
      ws[OFF_KK + srow * 128 + ccol] = acc[v];
    }
  }
}

// ---------------------------------------------------------------------------
// K4 (WMMA): chunk injections r[(b,m)][n] = sum_j u[b, m*Q+j] * vs[Q-1-j][n]
//            M=512 (b*32+m), K=128, N=64
// ---------------------------------------------------------------------------
__global__ __launch_bounds__(256) void ssm_R(const float* __restrict__ u,
                                             float* __restrict__ ws) {
  const int tid  = threadIdx.x;
  const int wave = tid >> 5;
  const int lane = tid & 31;
  const int half = lane >> 4;
  const int lc   = lane & 15;
  const float* vs = ws + OFF_VS;

  const int gt = blockIdx.x * 8 + wave;       // 128 tiles total
  const int rt = gt >> 2;                     // 0..31 row tile (M)
  const int nt = gt & 3;                      // 0..3  col tile (N)

  v8f acc = {0.f, 0.f, 0.f, 0.f, 0.f, 0.f, 0.f, 0.f};
  #pragma unroll
  for (int kb = 0; kb < 32; ++kb) {           // K = 128
    const int j0 = kb * 4 + 2 * half;
    const int r  = rt * 16 + lc;
    const int bb = r >> 5, mm = r & 31;
    v2f a, b;
    a.x = u[bb * SEQ + mm * QCH + j0];
    a.y = u[bb * SEQ + mm * QCH + j0 + 1];
    const int n = nt * 16 + lc;
    b.x = vs[(QCH - 1 - j0) * 64 + n];
    b.y = vs[(QCH - 2 - j0) * 64 + n];
    acc = wmma_f32_k4(a, b, acc);
  }
  const int n = nt * 16 + lc;
  #pragma unroll
  for (int v = 0; v < 8; ++v) {
    const int r = rt * 16 + half * 8 + v;
    ws[OFF_R + r * 64 + n] = acc[v];
  }
}

// ---------------------------------------------------------------------------
// K5: chunk-boundary states x[0]=0; x[m+1] = P x[m] + r[m]  (31 steps)
// ---------------------------------------------------------------------------
__global__ __launch_bounds__(256) void ssm_states(float* __restrict__ ws) {
  __shared__ float P[4096];
  __shared__ float xa[1024];                  // 16 batches x 64
  __shared__ float xb[1024];
  const int tid = threadIdx.x;

  for (int idx = tid; idx < 4096; idx += 256) P[idx] = ws[OFF_P + idx];
  for (int idx = tid; idx < 1024; idx += 256) {
    xa[idx] = 0.0f;
    ws[OFF_X + idx] = 0.0f;                   // x[0] = 0
  }
  __syncthreads();

  for (int m = 0; m < NCHK - 1; ++m) {
    const float* cur = (m & 1) ? xb : xa;
    float*       nxt = (m & 1) ? xa : xb;
    for (int idx = tid; idx < 1024; idx += 256) {
      const int bb = idx >> 6, n = idx & 63;
      float s = ws[OFF_R + (bb * NCHK + m) * 64 + n];
      #pragma unroll 8
      for (int k = 0; k < 64; ++k) s += P[n * 64 + k] * cur[bb * 64 + k];
      nxt[idx] = s;
      ws[OFF_X + (m + 1) * 1024 + idx] = s;
    }
    __syncthreads();
  }
}

// ---------------------------------------------------------------------------
// K6 (main, WMMA + async LDS staging): one workgroup per (batch, chunk).
// Per 16x16 tile: acc = Toeplitz(u_chunk) @ k   (K=128, intra-chunk conv)
//                     + W @ C^T                 (K=64, state carry-in)
// then y = acc + D*u.  48 chained v_wmma_f32_16x16x4_f32 per tile.
// Operand staging uses global_load_async_to_lds_b128: A_bar is issued first,
// waited with s_wait_asynccnt 20 (in-order completion), the taps (64KB) and
// C (17KB) keep streaming while the sequential state chain runs.
//
// LDS layout (floats):
//   u_s  @     0 : 128                  u chunk
//   xv   @   128 : 64                   x_m
//   A_s  @   192 : 64 rows x 68 (pad)   A_bar   (async, 16B rows stride 272B)
//   w_s  @  4544 : 128 rows x 65 (pad)  state trajectory
//   C_s  @ 12864 : 64 rows x 68 (pad)   C[c][n] (async)
//   kk_s @ 17216 : 128 x 128            taps    (async, contiguous)
//   total 33600 floats = 134400 B of the 320KB WGP LDS
// ---------------------------------------------------------------------------
#define SM_U   0
#define SM_XV  128
#define SM_A   192
#define SM_W   4544
#define SM_C   12864
#define SM_KK  17216
#define SM_TOT 33600

__global__ __launch_bounds__(256) void ssm_main(const float* __restrict__ u,
                                                const float* __restrict__ C,
                                                const float* __restrict__ D,
                                                const float* __restrict__ ws,
                                                float* __restrict__ y) {
  extern __shared__ float sm[];
  float* u_s  = sm + SM_U;
  float* xv   = sm + SM_XV;
  float* A_s  = sm + SM_A;
  float* w_s  = sm + SM_W;
  float* C_s  = sm + SM_C;
  float* kk_s = sm + SM_KK;

  const int tid = threadIdx.x;
  const int b   = blockIdx.x >> 5;  // batch
  const int m   = blockIdx.x & 31;  // chunk

  // byte offset of dynamic LDS base (no static LDS in this kernel)
  const unsigned lds_base = (unsigned)__builtin_amdgcn_groupstaticsize();

  // ---- issue async global->LDS copies (16B per lane per instruction) ----
  // A_bar first: 1024 chunks -> 4 instructions per wave
  for (int idx = tid; idx < 1024; idx += 256) {
    const int r = idx >> 4, ch = idx & 15;
    async_ld_b128(lds_base + SM_A * 4 + r * 272 + ch * 16,
                  ws + OFF_AB + r * 64 + ch * 4);
  }
  // conv taps: 4096 chunks -> 16 instructions per wave
  for (int idx = tid; idx < 4096; idx += 256) {
    async_ld_b128(lds_base + SM_KK * 4 + idx * 16, ws + OFF_KK + idx * 4);
  }
  // C (row-major [c][n], padded rows): 1024 chunks -> 4 instructions per wave
  for (int idx = tid; idx < 1024; idx += 256) {
    const int r = idx >> 4, ch = idx & 15;
    async_ld_b128(lds_base + SM_C * 4 + r * 272 + ch * 16,
                  C + r * 64 + ch * 4);
  }

  // small operands via normal path
  if (tid < 128) u_s[tid] = u[b * SEQ + m * QCH + tid];
  if (tid >= 128 && tid < 192) {
    const int n = tid - 128;
    xv[n] = ws[OFF_X + m * 1024 + b * 64 + n];
  }

  // A_bar copies were the first 4 async ops of this wave; async loads retire
  // in order, so ASYNCcnt <= 20 guarantees A_s is in LDS (taps/C may still fly)
  asm volatile("s_wait_asynccnt 20" ::: "memory");
  __syncthreads();

  // ---- in-chunk state trajectory: w[i] = A_bar^{i+1} x_m (sequential),
  //      overlapped with the outstanding tap/C async copies ----
  for (int i = 0; i < QCH; ++i) {
    if (tid < 64) {
      const float* src = (i == 0) ? xv : &w_s[(i - 1) * 65];
      float s = 0.0f;
      #pragma unroll 8
      for (int k = 0; k < 64; ++k) s += A_s[tid * 68 + k] * src[k];
      w_s[i * 65 + tid] = s;
    }
    __syncthreads();
  }

  asm volatile("s_wait_asynccnt 0" ::: "memory");
  __syncthreads();

  // ---- WMMA tiles: 8 i-tiles x 8 c-tiles, one tile per wave-iteration ----
  const int wave = tid >> 5;
  const int lane = tid & 31;
  const int half = lane >> 4;
  const int lc   = lane & 15;

  for (int tt = wave; tt < 64; tt += 8) {
    const int it = tt >> 3, ct = tt & 7;
    const int i0 = it * 16, c0 = ct * 16;
    v8f acc = {0.f, 0.f, 0.f, 0.f, 0.f, 0.f, 0.f, 0.f};

    // intra-chunk causal conv: K = 128 over lag s (Toeplitz built on the fly)
    #pragma unroll
    for (int kb = 0; kb < 32; ++kb) {
      const int s0 = kb * 4 + 2 * half;
      const int mr = i0 + lc;
      v2f a, bfr;
      a.x = (mr >= s0)     ? u_s[mr - s0]     : 0.0f;
      a.y = (mr >= s0 + 1) ? u_s[mr - s0 - 1] : 0.0f;
      bfr.x = kk_s[s0 * 128 + c0 + lc];
      bfr.y = kk_s[(s0 + 1) * 128 + c0 + lc];
      acc = wmma_f32_k4(a, bfr, acc);
    }
    // state carry-in: K = 64 over state dim n; B[n][c] = C_s[c*68+n]
    #pragma unroll
    for (int kb = 0; kb < 16; ++kb) {
      const int n0 = kb * 4 + 2 * half;
      v2f a, bfr;
      a.x = w_s[(i0 + lc) * 65 + n0];
      a.y = w_s[(i0 + lc) * 65 + n0 + 1];
      bfr.x = C_s[(c0 + lc) * 68 + n0];
      bfr.y = C_s[(c0 + lc) * 68 + n0 + 1];
      acc = wmma_f32_k4(a, bfr, acc);
    }

    // epilogue: + D*u skip, two 16B stores per lane (contiguous along t)
    const int c  = c0 + lc;
    const int ib = i0 + half * 8;
    const float d = D[c];
    float4 o0, o1;
    o0.x = acc[0] + d * u_s[ib + 0];
    o0.y = acc[1] + d * u_s[ib + 1];
    o0.z = acc[2] + d * u_s[ib + 2];
    o0.w = acc[3] + d * u_s[ib + 3];
    o1.x = acc[4] + d * u_s[ib + 4];
    o1.y = acc[5] + d * u_s[ib + 5];
    o1.z = acc[6] + d * u_s[ib + 6];
    o1.w = acc[7] + d * u_s[ib + 7];
    const size_t base = ((size_t)b * CH + c) * SEQ + (size_t)m * QCH + ib;
    *(float4*)(y + base)     = o0;
    *(float4*)(y + base + 4) = o1;
  }
}

// ---------------------------------------------------------------------------
extern "C" void kernel_launch(void* const* d_in, const int* in_sizes, int n_in,
                              void* d_out, int out_size, void* d_ws, size_t ws_size,
                              hipStream_t stream) {
  const float* u  = (const float*)d_in[0];   // (16, 4096)
  const float* A  = (const float*)d_in[1];   // (64, 64)
  const float* Bv = (const float*)d_in[2];   // (64, 1)
  const float* C  = (const float*)d_in[3];   // (128, 64)
  const float* D  = (const float*)d_in[4];   // (128, 1)
  const float* dt = (const float*)d_in[5];   // (1,)
  float* y  = (float*)d_out;                 // (16, 128, 4096)
  float* ws = (float*)d_ws;                  // ~394 KB used

  (void)in_sizes; (void)n_in; (void)out_size; (void)ws_size;

  const size_t sh_main = SM_TOT * sizeof(float);  // ~131 KB of 320 KB WGP LDS
  // hipFuncSetAttribute is not a stream op; safe under graph capture.
  (void)hipFuncSetAttribute((const void*)ssm_main,
                            hipFuncAttributeMaxDynamicSharedMemorySize,
                            (int)sh_main);

  ssm_setup <<<1,   256, 0,       stream>>>(A, Bv, dt, ws);
  ssm_vs_pow<<<1,   256, 0,       stream>>>(ws);
  ssm_kk    <<<1,   256, 0,       stream>>>(C, ws);
  ssm_R     <<<16,  256, 0,       stream>>>(u, ws);
  ssm_states<<<1,   256, 0,       stream>>>(ws);
  ssm_main  <<<512, 256, sh_main, stream>>>(u, C, D, ws, y);
}